// AttentionMPLayer_66537633349677
// MI455X (gfx1250) — compile-verified
//
#include <hip/hip_runtime.h>
#include <math.h>

#define H 48
#define EPS_LN 1e-5f

typedef __attribute__((ext_vector_type(2))) float v2f;
typedef __attribute__((ext_vector_type(8))) float v8f;

// ---------------------------------------------------------------------------
// float atomic max via int/uint monotone trick (lowers to global_atomic ops)
// ---------------------------------------------------------------------------
__device__ inline void atomicMaxFloat(float* addr, float v) {
    if (v >= 0.0f) atomicMax((int*)addr, __float_as_int(v));
    else           atomicMin((unsigned int*)addr, __float_as_uint(v));
}

// ---------------------------------------------------------------------------
// Kernel A: P = h @ W^T  (mat 0: Q with LN(gq,bq), 1: K with LN(gk,bk), 2: M raw)
// grid = (N/16, 3), block = 96 (3 waves). Wave w computes 16x16 col-tile w.
// WMMA f32 16x16x4: A lane layout: row = lane&15, K-pair = 2*(lane>>4);
// B lane layout: col = lane&15, K-pair = 2*(lane>>4); B[k][n] = W[n][k].
// D: lane holds col (lane&15), rows (lane>>4)*8 .. +7 in v[0..7].
// ---------------------------------------------------------------------------
__global__ __launch_bounds__(96)
void node_gemm_ln(const float* __restrict__ h,
                  const float* __restrict__ Wq, const float* __restrict__ Wk,
                  const float* __restrict__ Wm,
                  const float* __restrict__ gq, const float* __restrict__ bq,
                  const float* __restrict__ gk, const float* __restrict__ bk,
                  float* __restrict__ Q, float* __restrict__ K,
                  float* __restrict__ M)
{
    __shared__ float tile[16][H + 2];
    const int mat = blockIdx.y;
    const float* W = (mat == 0) ? Wq : (mat == 1) ? Wk : Wm;
    float* out     = (mat == 0) ? Q  : (mat == 1) ? K  : M;

    const int node0 = blockIdx.x * 16;
    const int lane  = threadIdx.x & 31;
    const int wave  = threadIdx.x >> 5;          // col tile 0..2
    const int row   = lane & 15;
    const int koff  = (lane >> 4) * 2;
    const int ncol  = wave * 16 + row;

    v8f acc = {};
    #pragma unroll
    for (int k0 = 0; k0 < H; k0 += 4) {
        const float* ap = h + (size_t)(node0 + row) * H + k0 + koff;
        const float* bp = W + (size_t)ncol * H + k0 + koff;
        v2f a; a[0] = ap[0]; a[1] = ap[1];
        v2f b; b[0] = bp[0]; b[1] = bp[1];
        acc = __builtin_amdgcn_wmma_f32_16x16x4_f32(false, a, false, b,
                                                    (short)0, acc, false, false);
    }

    const int rbase = (lane >> 4) * 8;
    if (mat == 2) {                              // M: no layernorm, store direct
        #pragma unroll
        for (int r = 0; r < 8; ++r)
            out[(size_t)(node0 + rbase + r) * H + ncol] = acc[r];
        return;
    }

    #pragma unroll
    for (int r = 0; r < 8; ++r) tile[rbase + r][ncol] = acc[r];
    __syncthreads();

    const float* g = (mat == 0) ? gq : gk;
    const float* b = (mat == 0) ? bq : bk;
    if (threadIdx.x < 16) {
        const int t = threadIdx.x;
        float mu = 0.0f;
        for (int c = 0; c < H; ++c) mu += tile[t][c];
        mu *= (1.0f / H);
        float var = 0.0f;
        for (int c = 0; c < H; ++c) { float d = tile[t][c] - mu; var += d * d; }
        var *= (1.0f / H);
        const float rs = rsqrtf(var + EPS_LN);
        float* op = out + (size_t)(node0 + t) * H;
        for (int c = 0; c < H; ++c)
            op[c] = (tile[t][c] - mu) * rs * g[c] + b[c];
    }
}

// ---------------------------------------------------------------------------
// Init: smax=-inf, ssum=0, logmult=log(max(nm,1)), agg=0
// ---------------------------------------------------------------------------
__global__ void init_kernel(const float* __restrict__ nm,
                            float* __restrict__ smax, float* __restrict__ ssum,
                            float* __restrict__ lmul, float* __restrict__ agg,
                            int N)
{
    const int i = blockIdx.x * blockDim.x + threadIdx.x;
    if (i < N) {
        smax[i] = -INFINITY;
        ssum[i] = 0.0f;
        const float m = nm[i];
        lmul[i] = logf(m > 1.0f ? m : 1.0f);
    }
    if (i < N * H) agg[i] = 0.0f;
}

// ---------------------------------------------------------------------------
// Edge pass 1: score = Q[src]·(K[dst] + 0.1*ef) + log(clip(mult[dst],1));
// segment-max into smax[src].
// ---------------------------------------------------------------------------
__global__ __launch_bounds__(256)
void edge_score(const int* __restrict__ ei, const float* __restrict__ ea,
                const float* __restrict__ Q, const float* __restrict__ K,
                const float* __restrict__ lmul,
                float* __restrict__ score, float* __restrict__ smax, int E)
{
    const int e = blockIdx.x * blockDim.x + threadIdx.x;
    if (e >= E) return;
    const int src = ei[e];
    const int dst = ei[E + e];
    const float4* q = (const float4*)(Q + (size_t)src * H);
    const float4* k = (const float4*)(K + (size_t)dst * H);
    const float4* f = (const float4*)(ea + (size_t)e * H);
    float s = 0.0f;
    #pragma unroll
    for (int i = 0; i < H / 4; ++i) {
        const float4 qv = q[i], kv = k[i], fv = f[i];
        s += qv.x * (kv.x + 0.1f * fv.x);
        s += qv.y * (kv.y + 0.1f * fv.y);
        s += qv.z * (kv.z + 0.1f * fv.z);
        s += qv.w * (kv.w + 0.1f * fv.w);
    }
    s += lmul[dst];
    score[e] = s;
    atomicMaxFloat(smax + src, s);
}

// ---------------------------------------------------------------------------
// Edge pass 2: es = exp(min(score - smax[src],20));
// ssum[src] += es;  agg[src][:] += es * M[dst][:]   (unnormalized accumulate;
// the /(ssum+1e-10) at node level reproduces alpha-weighted sum exactly)
// ---------------------------------------------------------------------------
__global__ __launch_bounds__(256)
void edge_accum(const int* __restrict__ ei, const float* __restrict__ score,
                const float* __restrict__ smax, const float* __restrict__ Mh,
                float* __restrict__ ssum, float* __restrict__ agg, int E)
{
    const int e = blockIdx.x * blockDim.x + threadIdx.x;
    if (e >= E) return;
    const int src = ei[e];
    const int dst = ei[E + e];
    const float es = expf(fminf(score[e] - smax[src], 20.0f));
    atomicAdd(ssum + src, es);
    const float4* m4 = (const float4*)(Mh + (size_t)dst * H);
    float* a = agg + (size_t)src * H;
    #pragma unroll
    for (int i = 0; i < H / 4; ++i) {
        const float4 mv = m4[i];
        atomicAdd(a + 4 * i + 0, es * mv.x);
        atomicAdd(a + 4 * i + 1, es * mv.y);
        atomicAdd(a + 4 * i + 2, es * mv.z);
        atomicAdd(a + 4 * i + 3, es * mv.w);
    }
}

// ---------------------------------------------------------------------------
// Kernel C: z = [h, agg/(ssum+1e-10)] @ Wu^T ; out = LN(h + leaky_relu(z))
// grid = N/16, block = 96.  A is 16x96 (K loop 24 steps of 4).
// ---------------------------------------------------------------------------
__global__ __launch_bounds__(96)
void node_update(const float* __restrict__ h, const float* __restrict__ agg,
                 const float* __restrict__ ssum, const float* __restrict__ Wu,
                 const float* __restrict__ go, const float* __restrict__ bo,
                 float* __restrict__ out)
{
    __shared__ float ztile[16][H + 2];
    __shared__ float inv[16];
    const int node0 = blockIdx.x * 16;
    const int lane  = threadIdx.x & 31;
    const int wave  = threadIdx.x >> 5;
    const int row   = lane & 15;
    const int koff  = (lane >> 4) * 2;
    const int ncol  = wave * 16 + row;

    if (threadIdx.x < 16)
        inv[threadIdx.x] = 1.0f / (ssum[node0 + threadIdx.x] + 1e-10f);
    __syncthreads();

    v8f acc = {};
    #pragma unroll
    for (int k0 = 0; k0 < 2 * H; k0 += 4) {
        const int k = k0 + koff;                 // pairs never straddle k=48
        v2f a;
        if (k < H) {
            const float* ap = h + (size_t)(node0 + row) * H + k;
            a[0] = ap[0]; a[1] = ap[1];
        } else {
            const float* ap = agg + (size_t)(node0 + row) * H + (k - H);
            const float s = inv[row];
            a[0] = ap[0] * s; a[1] = ap[1] * s;
        }
        const float* bp = Wu + (size_t)ncol * (2 * H) + k;
        v2f b; b[0] = bp[0]; b[1] = bp[1];
        acc = __builtin_amdgcn_wmma_f32_16x16x4_f32(false, a, false, b,
                                                    (short)0, acc, false, false);
    }

    const int rbase = (lane >> 4) * 8;
    #pragma unroll
    for (int r = 0; r < 8; ++r) ztile[rbase + r][ncol] = acc[r];
    __syncthreads();

    if (threadIdx.x < 16) {
        const int t = threadIdx.x;
        const float* hp = h + (size_t)(node0 + t) * H;
        float y[H];
        float mu = 0.0f;
        for (int c = 0; c < H; ++c) {
            float z = ztile[t][c];
            z = (z > 0.0f) ? z : 0.01f * z;      // leaky_relu(0.01)
            const float v = hp[c] + z;           // residual
            y[c] = v; mu += v;
        }
        mu *= (1.0f / H);
        float var = 0.0f;
        for (int c = 0; c < H; ++c) { const float d = y[c] - mu; var += d * d; }
        var *= (1.0f / H);
        const float rs = rsqrtf(var + EPS_LN);
        float* op = out + (size_t)(node0 + t) * H;
        for (int c = 0; c < H; ++c)
            op[c] = (y[c] - mu) * rs * go[c] + bo[c];
    }
}

// ---------------------------------------------------------------------------
extern "C" void kernel_launch(void* const* d_in, const int* in_sizes, int n_in,
                              void* d_out, int out_size, void* d_ws, size_t ws_size,
                              hipStream_t stream)
{
    const float* h   = (const float*)d_in[0];
    const int*   ei  = (const int*)  d_in[1];
    const float* ea  = (const float*)d_in[2];
    const float* nm  = (const float*)d_in[3];
    const float* Wq  = (const float*)d_in[4];
    const float* Wk  = (const float*)d_in[5];
    const float* Wm  = (const float*)d_in[6];
    const float* Wu  = (const float*)d_in[7];
    const float* gq  = (const float*)d_in[8];
    const float* bq  = (const float*)d_in[9];
    const float* gk  = (const float*)d_in[10];
    const float* bk  = (const float*)d_in[11];
    const float* go  = (const float*)d_in[12];
    const float* bo  = (const float*)d_in[13];

    const int N = in_sizes[0] / H;   // 100000 (divisible by 16)
    const int E = in_sizes[1] / 2;   // 1600000

    float* ws = (float*)d_ws;
    size_t off = 0;
    float* Q     = ws + off; off += (size_t)N * H;
    float* K     = ws + off; off += (size_t)N * H;
    float* M     = ws + off; off += (size_t)N * H;
    float* agg   = ws + off; off += (size_t)N * H;
    float* smax  = ws + off; off += N;
    float* ssum  = ws + off; off += N;
    float* lmul  = ws + off; off += N;
    float* score = ws + off; off += E;

    hipLaunchKernelGGL(init_kernel, dim3((N * H + 255) / 256), dim3(256), 0, stream,
                       nm, smax, ssum, lmul, agg, N);
    hipLaunchKernelGGL(node_gemm_ln, dim3(N / 16, 3), dim3(96), 0, stream,
                       h, Wq, Wk, Wm, gq, bq, gk, bk, Q, K, M);
    hipLaunchKernelGGL(edge_score, dim3((E + 255) / 256), dim3(256), 0, stream,
                       ei, ea, Q, K, lmul, score, smax, E);
    hipLaunchKernelGGL(edge_accum, dim3((E + 255) / 256), dim3(256), 0, stream,
                       ei, score, smax, M, ssum, agg, E);
    hipLaunchKernelGGL(node_update, dim3(N / 16), dim3(96), 0, stream,
                       h, agg, ssum, Wu, go, bo, (float*)d_out);
}